// TriangSag_40733469835821
// MI455X (gfx1250) — compile-verified
//
#include <hip/hip_runtime.h>
#include <math.h>

#define TPB 256
static const int BGR = 10000;   // number of graphs (reference constant B)

typedef float v2f __attribute__((ext_vector_type(2)));
typedef float v8f __attribute__((ext_vector_type(8)));

// ---- monotone float <-> uint encoding for atomic max on floats ----
static __device__ __forceinline__ unsigned fenc(float f) {
  unsigned u = __float_as_uint(f);
  return (u & 0x80000000u) ? ~u : (u | 0x80000000u);
}
static __device__ __forceinline__ float fdec(unsigned u) {
  u = (u & 0x80000000u) ? (u & 0x7fffffffu) : ~u;
  return __uint_as_float(u);
}
#define SENT 0x007FFFFFu   // fenc(-inf)

// ======================= tiny utility kernels =======================
__global__ void k_fill_f32(float* __restrict__ p, float v, long long n) {
  long long i = (long long)blockIdx.x * blockDim.x + threadIdx.x;
  if (i < n) p[i] = v;
}
__global__ void k_fill_u32(unsigned* __restrict__ p, unsigned v, long long n) {
  long long i = (long long)blockIdx.x * blockDim.x + threadIdx.x;
  if (i < n) p[i] = v;
}
__global__ void k_copy_f32(float* __restrict__ d, const float* __restrict__ s, long long n) {
  long long i = (long long)blockIdx.x * blockDim.x + threadIdx.x;
  if (i < n) d[i] = s[i];
}
__global__ void k_pad_w1a(const float* __restrict__ W1a, float* __restrict__ Wpad) {
  int i = blockIdx.x * blockDim.x + threadIdx.x;   // 16*64 = 1024
  if (i < 16 * 64) Wpad[i] = (i < 15 * 64) ? W1a[i] : 0.0f;
}

// ======================= degree / one-hot (conv1 input) =======================
__global__ void k_deg_count(const int* __restrict__ src, int* __restrict__ cnt, int E) {
  int e = blockIdx.x * blockDim.x + threadIdx.x;
  if (e < E) atomicAdd(&cnt[src[e]], 1);
}
__global__ void k_onehot(float* __restrict__ H1, const int* __restrict__ cnt, int N) {
  long long i = (long long)blockIdx.x * blockDim.x + threadIdx.x;
  if (i < (long long)N * 16) {
    int n = (int)(i >> 4), c = (int)(i & 15);
    int d = cnt[n]; if (d > 14) d = 14;
    H1[i] = (c == d) ? 1.0f : 0.0f;
  }
}
// conv1 aggregation: x is one-hot -> just add emask at column deg[src]
__global__ void k_agg_onehot(const int* __restrict__ src, const int* __restrict__ dst,
                             const int* __restrict__ cnt, const float* __restrict__ emask,
                             float* __restrict__ H1, int E) {
  int e = blockIdx.x * blockDim.x + threadIdx.x;
  if (e >= E) return;
  float m = emask[e];
  if (m == 0.0f) return;
  int d = cnt[src[e]]; if (d > 14) d = 14;
  atomicAdd(&H1[(long long)dst[e] * 16 + d], m);
}
// generic 64-wide aggregation: h[dst] += x[src]*emask
__global__ void k_agg64(const int* __restrict__ src, const int* __restrict__ dst,
                        const float* __restrict__ emask,
                        const float* __restrict__ x, float* __restrict__ h, int E) {
  long long i = (long long)blockIdx.x * blockDim.x + threadIdx.x;
  if (i >= (long long)E * 64) return;
  int e = (int)(i >> 6), c = (int)(i & 63);
  float m = emask[e];
  if (m == 0.0f) return;
  atomicAdd(&h[(long long)dst[e] * 64 + c], x[(long long)src[e] * 64 + c] * m);
}

// ======================= fused GIN MLP via WMMA =======================
// out = relu( relu(H@Wa + ba) @ Wb + bb ) [* rowscale]
// One wave per 16-row tile. A fragments: lane m=L%16, K = 2*half + v.
// C/D layout: vgpr v holds row (v + 8*half), col = L%16 (+ 16*t).
__global__ __launch_bounds__(128)
void k_gin_mlp(const float* __restrict__ H, int K1,
               const float* __restrict__ Wa, const float* __restrict__ ba,
               const float* __restrict__ Wb, const float* __restrict__ bb,
               float* __restrict__ out, const float* __restrict__ rowscale, int N) {
  __shared__ float smem[4 * 16 * 64];
  const int wave = threadIdx.x >> 5;
  const int lane = threadIdx.x & 31;
  const int half = (lane >> 4) & 1;
  const int nn   = lane & 15;
  const int tile = blockIdx.x * 4 + wave;
  const int rowbase = tile * 16;
  float* my = &smem[wave * 1024];

  // ---- phase 1: H[rowbase:rowbase+16, 0:K1] @ Wa  -> 16x64 (relu, +ba) ----
  v8f acc[4];
  #pragma unroll
  for (int t = 0; t < 4; ++t) acc[t] = (v8f){0,0,0,0,0,0,0,0};

  const int arow = rowbase + nn;
  for (int k0 = 0; k0 < K1; k0 += 4) {
    v2f a = (v2f){0.0f, 0.0f};
    if (arow < N) a = *(const v2f*)(H + (long long)arow * K1 + k0 + 2 * half);
    const float* wr0 = Wa + (long long)(k0 + 2 * half) * 64;
    const float* wr1 = wr0 + 64;
    #pragma unroll
    for (int t = 0; t < 4; ++t) {
      v2f b; b[0] = wr0[t * 16 + nn]; b[1] = wr1[t * 16 + nn];
      acc[t] = __builtin_amdgcn_wmma_f32_16x16x4_f32(
          false, a, false, b, (short)0, acc[t], false, false);
    }
  }
  #pragma unroll
  for (int t = 0; t < 4; ++t) {
    #pragma unroll
    for (int v = 0; v < 8; ++v) {
      float val = acc[t][v] + ba[t * 16 + nn];
      my[(v + 8 * half) * 64 + t * 16 + nn] = fmaxf(val, 0.0f);
    }
  }
  __syncthreads();

  // ---- phase 2: (16x64 intermediate) @ Wb -> 16x64 (relu, +bb) ----
  v8f d[4];
  #pragma unroll
  for (int t = 0; t < 4; ++t) d[t] = (v8f){0,0,0,0,0,0,0,0};
  for (int k0 = 0; k0 < 64; k0 += 4) {
    v2f a = *(const v2f*)&my[nn * 64 + k0 + 2 * half];
    const float* wr0 = Wb + (long long)(k0 + 2 * half) * 64;
    const float* wr1 = wr0 + 64;
    #pragma unroll
    for (int t = 0; t < 4; ++t) {
      v2f b; b[0] = wr0[t * 16 + nn]; b[1] = wr1[t * 16 + nn];
      d[t] = __builtin_amdgcn_wmma_f32_16x16x4_f32(
          false, a, false, b, (short)0, d[t], false, false);
    }
  }
  #pragma unroll
  for (int t = 0; t < 4; ++t) {
    #pragma unroll
    for (int v = 0; v < 8; ++v) {
      int r = rowbase + v + 8 * half;
      if (r < N) {
        float val = fmaxf(d[t][v] + bb[t * 16 + nn], 0.0f);
        if (rowscale) val *= rowscale[r];
        out[(long long)r * 64 + t * 16 + nn] = val;
      }
    }
  }
}

// ======================= GCN score =======================
__global__ void k_xw(const float* __restrict__ x, const float* __restrict__ W,
                     float* __restrict__ xw, int N) {
  int n = blockIdx.x * blockDim.x + threadIdx.x;
  if (n >= N) return;
  const float4* xp = (const float4*)(x + (long long)n * 64);
  const float4* wp = (const float4*)W;
  float s = 0.0f;
  #pragma unroll
  for (int i = 0; i < 16; ++i) {
    float4 a = xp[i], b = wp[i];
    s += a.x * b.x + a.y * b.y + a.z * b.z + a.w * b.w;
  }
  xw[n] = s;
}
__global__ void k_deg_edge(const int* __restrict__ dst, const float* __restrict__ emask,
                           float* __restrict__ degE, int E) {
  int e = blockIdx.x * blockDim.x + threadIdx.x;
  if (e >= E) return;
  float m = emask[e];
  if (m != 0.0f) atomicAdd(&degE[dst[e]], m);
}
__global__ void k_gcn_edge(const int* __restrict__ src, const int* __restrict__ dst,
                           const float* __restrict__ emask, const float* __restrict__ degE,
                           const float* __restrict__ xw, float* __restrict__ s, int E) {
  int e = blockIdx.x * blockDim.x + threadIdx.x;
  if (e >= E) return;
  float m = emask[e];
  if (m == 0.0f) return;
  int si = src[e], di = dst[e];
  float ds_ = degE[si], dd = degE[di];
  float is = (ds_ > 0.0f) ? rsqrtf(ds_) : 0.0f;
  float id = (dd > 0.0f) ? rsqrtf(dd) : 0.0f;
  atomicAdd(&s[di], xw[si] * is * id * m);
}
__global__ void k_gcn_self(const float* __restrict__ degE, const float* __restrict__ xw,
                           const float* __restrict__ nmask, const float* __restrict__ bg,
                           float* __restrict__ s, int N) {
  int n = blockIdx.x * blockDim.x + threadIdx.x;
  if (n >= N) return;
  float d = degE[n];
  float inv = (d > 0.0f) ? (1.0f / d) : 0.0f;
  s[n] = s[n] + xw[n] * inv * nmask[n] + bg[0];
}

// ======================= per-graph softmax + pooling =======================
__global__ void k_segmax(const float* __restrict__ s, const int* __restrict__ batch,
                         const float* __restrict__ nmask, unsigned* __restrict__ gmax, int N) {
  int n = blockIdx.x * blockDim.x + threadIdx.x;
  if (n >= N) return;
  if (nmask[n] > 0.0f) atomicMax(&gmax[batch[n]], fenc(s[n]));
}
__global__ void k_exp(float* __restrict__ s, const int* __restrict__ batch,
                      const float* __restrict__ nmask, const unsigned* __restrict__ gmax,
                      float* __restrict__ zsum, int N) {
  int n = blockIdx.x * blockDim.x + threadIdx.x;
  if (n >= N) return;
  unsigned gu = gmax[batch[n]];
  float m = (gu == SENT) ? 0.0f : fdec(gu);
  float e = (nmask[n] > 0.0f) ? expf(s[n] - m) : 0.0f;
  s[n] = e;
  if (e != 0.0f) atomicAdd(&zsum[batch[n]], e);
}
__global__ void k_pool_fin(float* __restrict__ s, const int* __restrict__ batch,
                           float* __restrict__ nmask, const float* __restrict__ zsum, int N) {
  int n = blockIdx.x * blockDim.x + threadIdx.x;
  if (n >= N) return;
  float ss = s[n] / (zsum[batch[n]] + 1e-16f);
  s[n] = ss;
  nmask[n] = nmask[n] * ((ss > 1e-3f) ? 1.0f : 0.0f);
}
__global__ void k_scale_x(float* __restrict__ x, const float* __restrict__ s,
                          const float* __restrict__ nmask, int N) {
  long long i = (long long)blockIdx.x * blockDim.x + threadIdx.x;
  if (i >= (long long)N * 64) return;
  int n = (int)(i >> 6);
  x[i] *= s[n] * nmask[n];
}
__global__ void k_emask_upd(const int* __restrict__ src, const int* __restrict__ dst,
                            const float* __restrict__ nmask, float* __restrict__ emask, int E) {
  int e = blockIdx.x * blockDim.x + threadIdx.x;
  if (e >= E) return;
  emask[e] *= nmask[src[e]] * nmask[dst[e]];
}

// ======================= global max pool + readout =======================
__global__ void k_gm_atomic(const float* __restrict__ pa, const int* __restrict__ batch,
                            const float* __restrict__ nmask, unsigned* __restrict__ gm, int N) {
  long long i = (long long)blockIdx.x * blockDim.x + threadIdx.x;
  if (i >= (long long)N * 64) return;
  int n = (int)(i >> 6), c = (int)(i & 63);
  if (nmask[n] > 0.0f) atomicMax(&gm[(long long)batch[n] * 64 + c], fenc(pa[i]));
}
__global__ void k_out_gemv(const unsigned* __restrict__ gm, const float* __restrict__ Wl,
                           const float* __restrict__ bl, float* __restrict__ out, int B) {
  int g = blockIdx.x * blockDim.x + threadIdx.x;
  if (g >= B) return;
  float acc = bl[0];
  #pragma unroll 8
  for (int c = 0; c < 64; ++c) {
    unsigned u = gm[(long long)g * 64 + c];
    float v = (u == SENT) ? 0.0f : fdec(u);
    acc += v * Wl[c];
  }
  out[g] = acc;
}

// ======================= attention + KL loss =======================
__global__ void k_attn_max(const float* __restrict__ x0, unsigned* __restrict__ encm, int N) {
  int n = blockIdx.x * blockDim.x + threadIdx.x;
  float v = (n < N) ? x0[n] : -INFINITY;
  #pragma unroll
  for (int o = 16; o > 0; o >>= 1) v = fmaxf(v, __shfl_xor(v, o));
  if ((threadIdx.x & 31) == 0 && v != -INFINITY) atomicMax(encm, fenc(v));
}
__global__ void k_attn_z(const float* __restrict__ x0, const unsigned* __restrict__ encm,
                         float* __restrict__ zall, int N) {
  int n = blockIdx.x * blockDim.x + threadIdx.x;
  float m = fdec(*encm);
  float e = (n < N) ? expf(x0[n] - m) : 0.0f;
  #pragma unroll
  for (int o = 16; o > 0; o >>= 1) e += __shfl_xor(e, o);
  if ((threadIdx.x & 31) == 0) atomicAdd(zall, e);
}
__global__ void k_loss_node(const float* __restrict__ x0, const int* __restrict__ batch,
                            const float* __restrict__ nmask, const float* __restrict__ s2,
                            const unsigned* __restrict__ encm, const float* __restrict__ zall,
                            float* __restrict__ esum, float* __restrict__ cnt, int N) {
  int n = blockIdx.x * blockDim.x + threadIdx.x;
  if (n >= N) return;
  float nm = nmask[n];
  if (nm <= 0.0f) return;
  float m = fdec(*encm);
  float attn = expf(x0[n] - m) / (*zall);
  float elem = attn * (logf(attn + 1e-30f) - logf(s2[n] + 1e-14f)) * nm;
  int g = batch[n];
  atomicAdd(&esum[g], elem);
  atomicAdd(&cnt[g], nm);
}
__global__ void k_loss_final(const float* __restrict__ esum, const float* __restrict__ cnt,
                             float* __restrict__ acc, int B) {
  int g = blockIdx.x * blockDim.x + threadIdx.x;
  float pg = 0.0f;
  if (g < B) {
    float c = cnt[g];
    pg = (c > 0.0f) ? (esum[g] / fmaxf(c, 1.0f)) : 0.0f;
    pg *= 100.0f;
  }
  #pragma unroll
  for (int o = 16; o > 0; o >>= 1) pg += __shfl_xor(pg, o);
  if ((threadIdx.x & 31) == 0) atomicAdd(acc, pg);
}
__global__ void k_loss_write(const float* __restrict__ acc, float* __restrict__ out, int B) {
  if (threadIdx.x == 0 && blockIdx.x == 0) out[0] = acc[0] / (float)B;
}

// ======================= host side =======================
static inline unsigned gridFor(long long n) { return (unsigned)((n + TPB - 1) / TPB); }

extern "C" void kernel_launch(void* const* d_in, const int* in_sizes, int n_in,
                              void* d_out, int out_size, void* d_ws, size_t ws_size,
                              hipStream_t stream) {
  const int N = in_sizes[0];
  const int E = in_sizes[1] / 2;
  const int B = BGR;

  const float* x0   = (const float*)d_in[0];
  const int*   eix  = (const int*)d_in[1];
  const int*   src  = eix;
  const int*   dst  = eix + E;
  const int*   batch = (const int*)d_in[2];
  const float* W1a = (const float*)d_in[3];  const float* b1a = (const float*)d_in[4];
  const float* W1b = (const float*)d_in[5];  const float* b1b = (const float*)d_in[6];
  const float* Wg1 = (const float*)d_in[7];  const float* bg1 = (const float*)d_in[8];
  const float* W2a = (const float*)d_in[9];  const float* b2a = (const float*)d_in[10];
  const float* W2b = (const float*)d_in[11]; const float* b2b = (const float*)d_in[12];
  const float* Wg2 = (const float*)d_in[13]; const float* bg2 = (const float*)d_in[14];
  const float* W3a = (const float*)d_in[15]; const float* b3a = (const float*)d_in[16];
  const float* W3b = (const float*)d_in[17]; const float* b3b = (const float*)d_in[18];
  const float* Wl  = (const float*)d_in[19]; const float* bl  = (const float*)d_in[20];

  float* out_logits = (float*)d_out;                 // [B]
  float* per_atom   = out_logits + B;                // [N,64]
  float* out_loss   = per_atom + (long long)N * 64;  // [1]

  // ---- bump allocator over workspace ----
  char* p = (char*)d_ws;
  auto alloc = [&](size_t bytes) -> void* {
    void* r = (void*)p;
    p += (bytes + 255) & ~(size_t)255;
    return r;
  };
  float*    bufA  = (float*)alloc((size_t)N * 64 * 4);  // x features
  float*    bufB  = (float*)alloc((size_t)N * 64 * 4);  // h = x + agg (prefix reused as [N,16] one-hot)
  float*    H1    = bufB;
  float*    WaPad = (float*)alloc(16 * 64 * 4);
  int*      degc  = (int*)alloc((size_t)N * 4);
  float*    s     = (float*)alloc((size_t)N * 4);
  float*    nmask = (float*)alloc((size_t)N * 4);
  float*    emask = (float*)alloc((size_t)E * 4);
  float*    degE  = (float*)alloc((size_t)N * 4);
  float*    xw    = (float*)alloc((size_t)N * 4);
  unsigned* gmax  = (unsigned*)alloc((size_t)B * 4);
  float*    zsum  = (float*)alloc((size_t)B * 4);
  unsigned* gm    = (unsigned*)alloc((size_t)B * 64 * 4);
  float*    esum  = (float*)alloc((size_t)B * 4);
  float*    cnt   = (float*)alloc((size_t)B * 4);
  unsigned* encm  = (unsigned*)alloc(4);
  float*    zall  = (float*)alloc(4);
  float*    lacc  = (float*)alloc(4);

  const long long N64 = (long long)N * 64;
  const long long E64 = (long long)E * 64;
  const int tiles = (N + 15) / 16;
  const unsigned mlpBlocks = (unsigned)((tiles + 3) / 4);

  // ---- init ----
  k_fill_u32<<<gridFor(N), TPB, 0, stream>>>((unsigned*)degc, 0u, N);
  k_fill_f32<<<gridFor(N), TPB, 0, stream>>>(nmask, 1.0f, N);
  k_fill_f32<<<gridFor(E), TPB, 0, stream>>>(emask, 1.0f, E);
  k_pad_w1a<<<4, TPB, 0, stream>>>(W1a, WaPad);

  // ---- degree one-hot features ----
  k_deg_count<<<gridFor(E), TPB, 0, stream>>>(src, degc, E);
  k_onehot<<<gridFor((long long)N * 16), TPB, 0, stream>>>(H1, degc, N);

  // ---- conv1 ----
  k_agg_onehot<<<gridFor(E), TPB, 0, stream>>>(src, dst, degc, emask, H1, E);
  k_gin_mlp<<<mlpBlocks, 128, 0, stream>>>(H1, 16, WaPad, b1a, W1b, b1b, bufA, nullptr, N);

  for (int pool = 0; pool < 2; ++pool) {
    const float* Wg = pool ? Wg2 : Wg1;
    const float* bg = pool ? bg2 : bg1;
    // GCN score
    k_xw<<<gridFor(N), TPB, 0, stream>>>(bufA, Wg, xw, N);
    k_copy_f32<<<gridFor(N), TPB, 0, stream>>>(degE, nmask, N);
    k_deg_edge<<<gridFor(E), TPB, 0, stream>>>(dst, emask, degE, E);
    k_fill_f32<<<gridFor(N), TPB, 0, stream>>>(s, 0.0f, N);
    k_gcn_edge<<<gridFor(E), TPB, 0, stream>>>(src, dst, emask, degE, xw, s, E);
    k_gcn_self<<<gridFor(N), TPB, 0, stream>>>(degE, xw, nmask, bg, s, N);
    // per-graph softmax
    k_fill_u32<<<gridFor(B), TPB, 0, stream>>>(gmax, SENT, B);
    k_fill_f32<<<gridFor(B), TPB, 0, stream>>>(zsum, 0.0f, B);
    k_segmax<<<gridFor(N), TPB, 0, stream>>>(s, batch, nmask, gmax, N);
    k_exp<<<gridFor(N), TPB, 0, stream>>>(s, batch, nmask, gmax, zsum, N);
    k_pool_fin<<<gridFor(N), TPB, 0, stream>>>(s, batch, nmask, zsum, N);
    // apply pooling
    k_scale_x<<<gridFor(N64), TPB, 0, stream>>>(bufA, s, nmask, N);
    k_emask_upd<<<gridFor(E), TPB, 0, stream>>>(src, dst, nmask, emask, E);
    // next conv (conv2 after pool1, conv3 after pool2)
    k_copy_f32<<<gridFor(N64), TPB, 0, stream>>>(bufB, bufA, N64);
    k_agg64<<<gridFor(E64), TPB, 0, stream>>>(src, dst, emask, bufA, bufB, E);
    if (pool == 0) {
      k_gin_mlp<<<mlpBlocks, 128, 0, stream>>>(bufB, 64, W2a, b2a, W2b, b2b, bufA, nullptr, N);
    } else {
      k_gin_mlp<<<mlpBlocks, 128, 0, stream>>>(bufB, 64, W3a, b3a, W3b, b3b, per_atom, nmask, N);
    }
  }

  // ---- global max pool + readout ----
  k_fill_u32<<<gridFor((long long)B * 64), TPB, 0, stream>>>(gm, SENT, (long long)B * 64);
  k_gm_atomic<<<gridFor(N64), TPB, 0, stream>>>(per_atom, batch, nmask, gm, N);
  k_out_gemv<<<gridFor(B), TPB, 0, stream>>>(gm, Wl, bl, out_logits, B);

  // ---- attention softmax (global) + KL loss ----
  k_fill_u32<<<1, 1, 0, stream>>>(encm, SENT, 1);
  k_fill_f32<<<1, 1, 0, stream>>>(zall, 0.0f, 1);
  k_fill_f32<<<1, 1, 0, stream>>>(lacc, 0.0f, 1);
  k_fill_f32<<<gridFor(B), TPB, 0, stream>>>(esum, 0.0f, B);
  k_fill_f32<<<gridFor(B), TPB, 0, stream>>>(cnt, 0.0f, B);
  k_attn_max<<<gridFor(N), TPB, 0, stream>>>(x0, encm, N);
  k_attn_z<<<gridFor(N), TPB, 0, stream>>>(x0, encm, zall, N);
  k_loss_node<<<gridFor(N), TPB, 0, stream>>>(x0, batch, nmask, s, encm, zall, esum, cnt, N);
  k_loss_final<<<gridFor(B), TPB, 0, stream>>>(esum, cnt, lacc, B);
  k_loss_write<<<1, 1, 0, stream>>>(lacc, out_loss, B);
}